// CPosModelBertIntent_19782619365729
// MI455X (gfx1250) — compile-verified
//
#include <hip/hip_runtime.h>
#include <hip/hip_bf16.h>
#include <math.h>

// ---------------------------------------------------------------------------
// CDNA5 (gfx1250) implementation of CPosModelBertIntent reference.
// Every GEMM uses V_WMMA_F32_16X16X4_F32 (native fp32 WMMA, wave32).
// Each wave computes FOUR 16x16 output tiles sharing one A fragment; all
// fragment fetches are contiguous 8-byte (v2f) loads -> global_load_b64 /
// ds_load_b64 with immediate offsets.
//
// VGPR layouts per CDNA5 ISA 7.12.2:
//  A (16x4 f32, MxK): lane<16 -> M=lane, regs {K0,K1}; lane>=16 -> M=lane-16, {K2,K3}
//  B (4x16 f32, KxN): lane<16 -> N=lane, regs {K0,K1}; lane>=16 -> N=lane-16, {K2,K3}
//  C/D (16x16 f32):   reg v: lane<16 -> (M=v,   N=lane); lane>=16 -> (M=v+8, N=lane-16)
// ---------------------------------------------------------------------------

typedef __attribute__((ext_vector_type(2))) float v2f;
typedef __attribute__((ext_vector_type(8))) float v8f;

#define B_   32
#define L_   512
#define D_   768
#define HATT 256
#define DV   768
#define E_   64
#define HL   512
#define NOUT 256
#define T_   12416
#define PADROWS (B_ * L_)   // 16384

__device__ __forceinline__ v8f wmma4(v2f a, v2f b, v8f c) {
  return __builtin_amdgcn_wmma_f32_16x16x4_f32(false, a, false, b, (short)0, c,
                                               false, false);
}
__device__ __forceinline__ v2f ld2(const float* p) { return *(const v2f*)p; }
__device__ __forceinline__ float sigmoidf_(float x) {
  return 1.0f / (1.0f + __expf(-x));
}

// ---------------------------------------------------------------------------
// 0a) packed<->padded row maps from seq_lens (int64).
// ---------------------------------------------------------------------------
__global__ void init_map_kernel(const long long* __restrict__ seq_lens,
                                int* __restrict__ pad2packed,
                                int* __restrict__ packed2pad) {
  __shared__ int off[B_ + 1];
  if (threadIdx.x == 0) {
    int s = 0;
    for (int b = 0; b < B_; ++b) { off[b] = s; s += (int)seq_lens[b]; }
    off[B_] = s;
  }
  __syncthreads();
  for (int r = threadIdx.x; r < PADROWS; r += blockDim.x) {
    int b = r >> 9, pos = r & (L_ - 1);
    int len = off[b + 1] - off[b];
    if (pos < len) {
      int pk = off[b] + pos;
      pad2packed[r] = pk;
      packed2pad[pk] = r;
    } else {
      pad2packed[r] = -1;
    }
  }
}

// ---------------------------------------------------------------------------
// 0b) Zero row (for pad-row A fragments) + transpose Wq/Wk/Wv to [n][k] so
//     B fragments are contiguous-k loads.
// ---------------------------------------------------------------------------
__global__ void prep_kernel(const float* __restrict__ Wq,
                            const float* __restrict__ Wk,
                            const float* __restrict__ Wv,
                            float* __restrict__ WqT, float* __restrict__ WkT,
                            float* __restrict__ WvT, float* __restrict__ zrow) {
  int idx = blockIdx.x * blockDim.x + threadIdx.x;
  int stride = gridDim.x * blockDim.x;
  for (int i = idx; i < D_; i += stride) zrow[i] = 0.0f;
  for (int i = idx; i < D_ * HATT; i += stride) {
    int k = i / HATT, n = i % HATT;
    WqT[n * D_ + k] = Wq[i];
    WkT[n * D_ + k] = Wk[i];
  }
  for (int i = idx; i < D_ * DV; i += stride) {
    int k = i / DV, n = i % DV;
    WvT[(size_t)n * D_ + k] = Wv[i];
  }
}

// ---------------------------------------------------------------------------
// 1) QKV projections.  N fused to 1280 cols = 20 groups of 64 (4 tiles/wave).
//    Pad rows read the zero row -> bias-only output (faithful to reference).
//    V is written TRANSPOSED per batch: vt[b][n][pos], so the S@V stage gets
//    contiguous B fragments.
// ---------------------------------------------------------------------------
__global__ __launch_bounds__(256) void qkv_kernel(
    const float* __restrict__ x, const float* __restrict__ y,
    const float* __restrict__ WqT, const float* __restrict__ bq,
    const float* __restrict__ WkT, const float* __restrict__ bk,
    const float* __restrict__ WvT, const float* __restrict__ bv,
    const int* __restrict__ pad2packed, const float* __restrict__ zrow,
    float* __restrict__ qout, float* __restrict__ kout, float* __restrict__ vt) {
  int job = blockIdx.x * 8 + (threadIdx.x >> 5);
  int mt = job / 20, g = job % 20;
  int lane = threadIdx.x & 31, l = lane & 15, hi = lane >> 4, kh = hi * 2;
  int m0 = mt * 16;

  const float* src; const float* WT; const float* bias; int n0, kind;
  if (g < 4)       { src = x; WT = WqT; bias = bq; n0 = g * 64;        kind = 0; }
  else if (g < 8)  { src = y; WT = WkT; bias = bk; n0 = (g - 4) * 64;  kind = 1; }
  else             { src = y; WT = WvT; bias = bv; n0 = (g - 8) * 64;  kind = 2; }

  int pr = pad2packed[m0 + l];
  const float* arow = (pr >= 0) ? (src + (size_t)pr * D_) : zrow;
  arow += kh;
  const float* b0 = WT + (size_t)(n0 + l) * D_ + kh;
  const float* b1 = b0 + (size_t)16 * D_;
  const float* b2 = b0 + (size_t)32 * D_;
  const float* b3 = b0 + (size_t)48 * D_;

  v8f acc[4] = {};
#pragma unroll 8
  for (int k0 = 0; k0 < D_; k0 += 4) {
    v2f a = ld2(arow + k0);
    acc[0] = wmma4(a, ld2(b0 + k0), acc[0]);
    acc[1] = wmma4(a, ld2(b1 + k0), acc[1]);
    acc[2] = wmma4(a, ld2(b2 + k0), acc[2]);
    acc[3] = wmma4(a, ld2(b3 + k0), acc[3]);
  }

  if (kind < 2) {
    float* dst = (kind == 0) ? qout : kout;
    for (int t = 0; t < 4; ++t) {
      int n = n0 + t * 16 + l;
      float bs = bias[n];
      for (int v = 0; v < 8; ++v)
        dst[(size_t)(m0 + v + 8 * hi) * HATT + n] = acc[t][v] + bs;
    }
  } else {
    int bb = m0 >> 9, pos0 = (m0 & (L_ - 1)) + 8 * hi;
    for (int t = 0; t < 4; ++t) {
      int n = n0 + t * 16 + l;
      float bs = bias[n];
      float* col = vt + ((size_t)bb * DV + n) * L_ + pos0;
      for (int v = 0; v < 8; ++v) col[v] = acc[t][v] + bs;
    }
  }
}

// ---------------------------------------------------------------------------
// 2) Attention per (batch, 16-row q-tile): S = Q K^T (16x512) in LDS,
//    softmax rows then /16 (softmax-first per reference), attn = S @ V.
// ---------------------------------------------------------------------------
__global__ __launch_bounds__(256) void attn_kernel(
    const float* __restrict__ qp, const float* __restrict__ kp,
    const float* __restrict__ vt, float* __restrict__ attn) {
  __shared__ float S[16][L_];  // 32 KB
  int b = blockIdx.x >> 5;
  int qt = blockIdx.x & 31;
  int wave = threadIdx.x >> 5, lane = threadIdx.x & 31;
  int l = lane & 15, hi = lane >> 4, kh = hi * 2;
  int qrow0 = b * L_ + qt * 16;

  // ---- S = Q @ K^T : 8 groups of 4 n-tiles, one group per wave ----
  const float* arow = qp + (size_t)(qrow0 + l) * HATT + kh;
  {
    int n0 = wave * 64;
    const float* b0 = kp + (size_t)(b * L_ + n0 + l) * HATT + kh;
    const float* b1 = b0 + (size_t)16 * HATT;
    const float* b2 = b0 + (size_t)32 * HATT;
    const float* b3 = b0 + (size_t)48 * HATT;
    v8f acc[4] = {};
#pragma unroll 8
    for (int k0 = 0; k0 < HATT; k0 += 4) {
      v2f a = ld2(arow + k0);
      acc[0] = wmma4(a, ld2(b0 + k0), acc[0]);
      acc[1] = wmma4(a, ld2(b1 + k0), acc[1]);
      acc[2] = wmma4(a, ld2(b2 + k0), acc[2]);
      acc[3] = wmma4(a, ld2(b3 + k0), acc[3]);
    }
    for (int t = 0; t < 4; ++t)
      for (int v = 0; v < 8; ++v) S[v + 8 * hi][n0 + t * 16 + l] = acc[t][v];
  }
  __syncthreads();

  // ---- softmax per row, then divide by sqrt(HATT)=16 ----
  for (int r = wave; r < 16; r += 8) {
    float m = -INFINITY;
    for (int i = lane; i < L_; i += 32) m = fmaxf(m, S[r][i]);
    for (int off = 16; off; off >>= 1) m = fmaxf(m, __shfl_xor(m, off, 32));
    float s = 0.0f;
    for (int i = lane; i < L_; i += 32) {
      float e = __expf(S[r][i] - m);
      S[r][i] = e;
      s += e;
    }
    for (int off = 16; off; off >>= 1) s += __shfl_xor(s, off, 32);
    float inv = 1.0f / (s * 16.0f);
    for (int i = lane; i < L_; i += 32) S[r][i] *= inv;
  }
  __syncthreads();

  // ---- attn = S @ V : 12 groups of 4 n-tiles over 8 waves ----
  const float* srow = &S[l][kh];
  for (int g = wave; g < 12; g += 8) {
    int n0 = g * 64;
    const float* b0 = vt + ((size_t)b * DV + n0 + l) * L_ + kh;
    const float* b1 = b0 + (size_t)16 * L_;
    const float* b2 = b0 + (size_t)32 * L_;
    const float* b3 = b0 + (size_t)48 * L_;
    v8f acc[4] = {};
#pragma unroll 8
    for (int k0 = 0; k0 < L_; k0 += 4) {
      v2f a = ld2(srow + k0);
      acc[0] = wmma4(a, ld2(b0 + k0), acc[0]);
      acc[1] = wmma4(a, ld2(b1 + k0), acc[1]);
      acc[2] = wmma4(a, ld2(b2 + k0), acc[2]);
      acc[3] = wmma4(a, ld2(b3 + k0), acc[3]);
    }
    for (int t = 0; t < 4; ++t)
      for (int v = 0; v < 8; ++v)
        attn[(size_t)(qrow0 + v + 8 * hi) * DV + n0 + t * 16 + l] = acc[t][v];
  }
}

// ---------------------------------------------------------------------------
// 3) Input gates: gx = [attn | prev_emb] @ W_ih^T + b_ih + b_hh.
//    K split: [0,768) from attn rows, [768,832) from prev-label embedding.
//    Pad rows use the zero row (lstm_in is scatter-into-zeros in reference).
// ---------------------------------------------------------------------------
__global__ __launch_bounds__(256) void gates_kernel(
    const float* __restrict__ attn, const float* __restrict__ Wih,
    const float* __restrict__ bih, const float* __restrict__ bhh,
    const float* __restrict__ emb, const float* __restrict__ init_t,
    const int* __restrict__ forced, const int* __restrict__ pad2packed,
    const long long* __restrict__ seq_lens, const float* __restrict__ zrow,
    float* __restrict__ gx) {
  int job = blockIdx.x * 8 + (threadIdx.x >> 5);
  int mt = job >> 5;           // / 32 groups
  int g = job & 31;
  int m0 = mt * 16, n0 = g * 64;
  int lane = threadIdx.x & 31, l = lane & 15, hi = lane >> 4, kh = hi * 2;

  int row = m0 + l;
  int bb = row >> 9, pos = row & (L_ - 1);
  bool valid = pos < (int)seq_lens[bb];
  const float* arow = (valid ? attn + (size_t)row * DV : zrow) + kh;
  const float* prow;
  if (!valid) prow = zrow;
  else if (pos == 0) prow = init_t;
  else {
    int pk = pad2packed[row];  // packed idx of (bb,pos); prev token = pk-1
    prow = emb + (size_t)forced[pk - 1] * E_;
  }
  prow += kh;

  const int LDW = DV + E_;  // 832
  const float* b0 = Wih + (size_t)(n0 + l) * LDW + kh;
  const float* b1 = b0 + (size_t)16 * LDW;
  const float* b2 = b0 + (size_t)32 * LDW;
  const float* b3 = b0 + (size_t)48 * LDW;

  v8f acc[4] = {};
#pragma unroll 8
  for (int k0 = 0; k0 < DV; k0 += 4) {
    v2f a = ld2(arow + k0);
    acc[0] = wmma4(a, ld2(b0 + k0), acc[0]);
    acc[1] = wmma4(a, ld2(b1 + k0), acc[1]);
    acc[2] = wmma4(a, ld2(b2 + k0), acc[2]);
    acc[3] = wmma4(a, ld2(b3 + k0), acc[3]);
  }
#pragma unroll
  for (int k0 = 0; k0 < E_; k0 += 4) {
    v2f a = ld2(prow + k0);
    acc[0] = wmma4(a, ld2(b0 + DV + k0), acc[0]);
    acc[1] = wmma4(a, ld2(b1 + DV + k0), acc[1]);
    acc[2] = wmma4(a, ld2(b2 + DV + k0), acc[2]);
    acc[3] = wmma4(a, ld2(b3 + DV + k0), acc[3]);
  }
  for (int t = 0; t < 4; ++t) {
    int n = n0 + t * 16 + l;
    float bias = bih[n] + bhh[n];
    for (int v = 0; v < 8; ++v)
      gx[(size_t)(m0 + v + 8 * hi) * (4 * HL) + n] = acc[t][v] + bias;
  }
}

// ---------------------------------------------------------------------------
// 4) Sequential LSTM: persistent single workgroup, 32 waves; h (64 KB) in LDS;
//    wave w owns hidden cols [16w,16w+16) plus matched i/f/g/o tiles -> cell
//    update fully in-register, c never leaves VGPRs.
// ---------------------------------------------------------------------------
__global__ __launch_bounds__(1024) void lstm_kernel(
    const float* __restrict__ gx, const float* __restrict__ Whh,
    float* __restrict__ hs) {
  extern __shared__ float hbuf[];  // B_ * HL floats = 64 KB
  int tid = threadIdx.x;
  int wave = tid >> 5, lane = tid & 31, l = lane & 15, hi = lane >> 4, kh = hi * 2;

  for (int i = tid; i < B_ * HL; i += 1024) hbuf[i] = 0.0f;
  __syncthreads();

  int j0 = wave * 16;
  const float* bi = Whh + (size_t)(j0 + l) * HL + kh;
  const float* bf = bi + (size_t)HL * HL;
  const float* bg = bf + (size_t)HL * HL;
  const float* bo = bg + (size_t)HL * HL;
  const float* h0row = hbuf + l * HL + kh;
  const float* h1row = hbuf + (16 + l) * HL + kh;

  float c[2][8];
  for (int mt = 0; mt < 2; ++mt)
    for (int v = 0; v < 8; ++v) c[mt][v] = 0.0f;

  for (int t = 0; t < L_; ++t) {
    v8f ai0 = {}, af0 = {}, ag0 = {}, ao0 = {};
    v8f ai1 = {}, af1 = {}, ag1 = {}, ao1 = {};
#pragma unroll 4
    for (int k0 = 0; k0 < HL; k0 += 4) {
      v2f bvi = ld2(bi + k0), bvf = ld2(bf + k0);
      v2f bvg = ld2(bg + k0), bvo = ld2(bo + k0);
      v2f a0 = ld2(h0row + k0), a1 = ld2(h1row + k0);
      ai0 = wmma4(a0, bvi, ai0);  ai1 = wmma4(a1, bvi, ai1);
      af0 = wmma4(a0, bvf, af0);  af1 = wmma4(a1, bvf, af1);
      ag0 = wmma4(a0, bvg, ag0);  ag1 = wmma4(a1, bvg, ag1);
      ao0 = wmma4(a0, bvo, ao0);  ao1 = wmma4(a1, bvo, ao1);
    }
    __syncthreads();  // all hbuf reads for step t complete

    for (int mt = 0; mt < 2; ++mt) {
      const v8f& ai = mt ? ai1 : ai0;
      const v8f& af = mt ? af1 : af0;
      const v8f& ag = mt ? ag1 : ag0;
      const v8f& ao = mt ? ao1 : ao0;
      for (int v = 0; v < 8; ++v) {
        int bb = mt * 16 + v + 8 * hi;
        int j = j0 + l;
        size_t grow = ((size_t)bb * L_ + t) * (4 * HL);
        float gi = ai[v] + gx[grow + j];
        float gf = af[v] + gx[grow + j + HL];
        float gg = ag[v] + gx[grow + j + 2 * HL];
        float go = ao[v] + gx[grow + j + 3 * HL];
        float cn = sigmoidf_(gf) * c[mt][v] + sigmoidf_(gi) * tanhf(gg);
        float hn = sigmoidf_(go) * tanhf(cn);
        c[mt][v] = cn;
        hbuf[bb * HL + j] = hn;
        hs[((size_t)bb * L_ + t) * HL + j] = hn;
      }
    }
    __syncthreads();  // new h visible before step t+1
  }
}

// ---------------------------------------------------------------------------
// 5) Output projection to packed rows: out = hs @ W_out^T + b_out.
//    T = 12416 = 776 * 16; 4 n-tile groups of 64 cols.
// ---------------------------------------------------------------------------
__global__ __launch_bounds__(256) void out_kernel(
    const float* __restrict__ hs, const float* __restrict__ Wout,
    const float* __restrict__ bout, const int* __restrict__ packed2pad,
    float* __restrict__ out) {
  int job = blockIdx.x * 8 + (threadIdx.x >> 5);
  int mt = job >> 2;
  int g = job & 3;
  int m0 = mt * 16, n0 = g * 64;
  int lane = threadIdx.x & 31, l = lane & 15, hi = lane >> 4, kh = hi * 2;

  int prow = packed2pad[m0 + l];
  const float* arow = hs + (size_t)prow * HL + kh;
  const float* b0 = Wout + (size_t)(n0 + l) * HL + kh;
  const float* b1 = b0 + (size_t)16 * HL;
  const float* b2 = b0 + (size_t)32 * HL;
  const float* b3 = b0 + (size_t)48 * HL;

  v8f acc[4] = {};
#pragma unroll 8
  for (int k0 = 0; k0 < HL; k0 += 4) {
    v2f a = ld2(arow + k0);
    acc[0] = wmma4(a, ld2(b0 + k0), acc[0]);
    acc[1] = wmma4(a, ld2(b1 + k0), acc[1]);
    acc[2] = wmma4(a, ld2(b2 + k0), acc[2]);
    acc[3] = wmma4(a, ld2(b3 + k0), acc[3]);
  }
  for (int t = 0; t < 4; ++t) {
    int n = n0 + t * 16 + l;
    float bias = bout[n];
    for (int v = 0; v < 8; ++v)
      out[(size_t)(m0 + v + 8 * hi) * NOUT + n] = acc[t][v] + bias;
  }
}

// ---------------------------------------------------------------------------
extern "C" void kernel_launch(void* const* d_in, const int* in_sizes, int n_in,
                              void* d_out, int out_size, void* d_ws, size_t ws_size,
                              hipStream_t stream) {
  const float* input_x  = (const float*)d_in[0];
  const float* input_y  = (const float*)d_in[1];
  const float* Wq       = (const float*)d_in[2];
  const float* bq       = (const float*)d_in[3];
  const float* Wk       = (const float*)d_in[4];
  const float* bk       = (const float*)d_in[5];
  const float* Wv       = (const float*)d_in[6];
  const float* bv       = (const float*)d_in[7];
  const float* emb      = (const float*)d_in[8];
  const float* init_t   = (const float*)d_in[9];
  const float* W_ih     = (const float*)d_in[10];
  const float* W_hh     = (const float*)d_in[11];
  const float* b_ih     = (const float*)d_in[12];
  const float* b_hh     = (const float*)d_in[13];
  const float* W_out    = (const float*)d_in[14];
  const float* b_out    = (const float*)d_in[15];
  const int*   forced   = (const int*)d_in[16];
  const long long* slen = (const long long*)d_in[17];
  float* out = (float*)d_out;

  // Workspace carve-up (floats).  Total ~306 MB.
  float* ws = (float*)d_ws;
  size_t o = 0;
  float* qp   = ws + o; o += (size_t)PADROWS * HATT;     // Q padded
  float* kp   = ws + o; o += (size_t)PADROWS * HATT;     // K padded
  float* vt   = ws + o; o += (size_t)B_ * DV * L_;       // V transposed [b][n][pos]
  float* attn = ws + o; o += (size_t)PADROWS * DV;
  float* gx   = ws + o; o += (size_t)PADROWS * 4 * HL;
  float* hs   = ws + o; o += (size_t)PADROWS * HL;
  float* WqT  = ws + o; o += (size_t)HATT * D_;
  float* WkT  = ws + o; o += (size_t)HATT * D_;
  float* WvT  = ws + o; o += (size_t)DV * D_;
  float* zrow = ws + o; o += D_;
  int* p2pk   = (int*)(ws + o); o += PADROWS;
  int* pk2p   = (int*)(ws + o); o += T_;

  init_map_kernel<<<1, 256, 0, stream>>>(slen, p2pk, pk2p);
  prep_kernel<<<512, 256, 0, stream>>>(Wq, Wk, Wv, WqT, WkT, WvT, zrow);

  // QKV: 1024 m-tiles x 20 groups = 20480 wave-jobs / 8 waves per block
  qkv_kernel<<<2560, 256, 0, stream>>>(input_x, input_y, WqT, bq, WkT, bk,
                                       WvT, bv, p2pk, zrow, qp, kp, vt);

  // Attention: 32 batches x 32 q-tiles
  attn_kernel<<<1024, 256, 0, stream>>>(qp, kp, vt, attn);

  // Input gates: 1024 m-tiles x 32 groups = 32768 wave-jobs
  gates_kernel<<<4096, 256, 0, stream>>>(attn, W_ih, b_ih, b_hh, emb, init_t,
                                         forced, p2pk, slen, zrow, gx);

  // Sequential LSTM: 1 workgroup, 32 waves, 64 KB dynamic LDS for h state
  lstm_kernel<<<1, 1024, B_ * HL * sizeof(float), stream>>>(gx, W_hh, hs);

  // Output projection: 776 m-tiles x 4 groups = 3104 wave-jobs
  out_kernel<<<388, 256, 0, stream>>>(hs, W_out, b_out, pk2p, out);
}